// SwarmLDA_26542897889837
// MI455X (gfx1250) — compile-verified
//
#include <hip/hip_runtime.h>

typedef __attribute__((ext_vector_type(2))) float v2f;
typedef __attribute__((ext_vector_type(8))) float v8f;

// Problem dims (fixed by the reference)
constexpr int B_ = 1024;      // cells
constexpr int G_ = 20000;     // genes
constexpr int L_ = 1000;      // LDAs
constexpr int g_ = 100;       // genes per LDA
constexpr int C_ = 10;        // cell types
constexpr int NPAD = 16;      // pad C to 16 for WMMA N

constexpr int TILES_M = B_ / 16;      // 64 row tiles
constexpr int KSPLIT  = 125;          // K segments per row tile
constexpr int KSEG    = G_ / KSPLIT;  // 160
constexpr int WSTEPS  = KSEG / 4;     // 40 wmma k-steps per wave
constexpr int UNROLL  = 8;            // k-steps batched per load clause
static_assert(WSTEPS % UNROLL == 0, "batching must divide steps");

// Weff pair-interleaved layout: W2(k, n) = (k>>1)*32 + n*2 + (k&1)
// -> the (k, k+1) pair a lane needs is one aligned float2.

// Workspace layout (in floats)
constexpr size_t OFF_WEFF  = 0;                                 // G_*NPAD = 320000
constexpr size_t OFF_STAGE = (size_t)G_ * NPAD;                 // TILES_M*256 = 16384
constexpr size_t OFF_BAGG  = OFF_STAGE + (size_t)TILES_M * 256; // 16
constexpr size_t WS_FLOATS = OFF_BAGG + 16;

__global__ void zero_ws_k(float* ws) {
    size_t i = (size_t)blockIdx.x * 256 + threadIdx.x;
    if (i < WS_FLOATS) ws[i] = 0.0f;
}

__global__ void bias_agg_k(const float* __restrict__ bias,
                           const float* __restrict__ agg_w,
                           const float* __restrict__ agg_b,
                           float* __restrict__ bagg) {
    int c = threadIdx.x;
    if (c < C_) {
        float s = agg_b[0];
        for (int l = 0; l < L_; ++l) s += agg_w[l] * bias[l * C_ + c];
        bagg[c] = s;
    }
}

// Fold agg_w into W and scatter-add into pair-interleaved Weff2
__global__ void scatter_k(const float* __restrict__ W,
                          const float* __restrict__ agg_w,
                          const int* __restrict__ gidx,
                          float* __restrict__ weff) {
    int t = blockIdx.x * 256 + threadIdx.x;
    if (t >= L_ * g_) return;
    int l = t / g_;
    int k = gidx[t];                 // 0..G-1
    float aw = agg_w[l];
    const float* w = W + (size_t)t * C_;
    float* dst = weff + (size_t)(k >> 1) * 32 + (k & 1);   // + n*2 per channel
#pragma unroll
    for (int c = 0; c < C_; ++c) atomicAdd(dst + c * 2, aw * w[c]);
}

// One wave per (row-tile, k-segment): D(16x16) += A(16x4,f32) x B(4x16,f32).
// A and B are both single b64 loads per step; batches of 8 steps per clause.
__global__ __launch_bounds__(32)
void gemm_k(const float* __restrict__ X,
            const float* __restrict__ weff,
            float* __restrict__ stage) {
    const int tile = blockIdx.x % TILES_M;
    const int seg  = blockIdx.x / TILES_M;
    const int lane = threadIdx.x;       // 0..31, wave32
    const int half = lane >> 4;         // 0: K{0,1}  1: K{2,3}
    const int sub  = lane & 15;         // A: row M; B: col N
    const int koff = half * 2;
    const int kbase = seg * KSEG;

    // Running pointers: immediate-offset loads inside each batch.
    const float* ap = X + (size_t)(tile * 16 + sub) * G_ + kbase + koff;      // +4 floats / step
    const float* bp = weff + (size_t)((kbase + koff) >> 1) * 32 + sub * 2;    // +64 floats / step

    v8f acc = {0.f, 0.f, 0.f, 0.f, 0.f, 0.f, 0.f, 0.f};

    for (int o = 0; o < WSTEPS / UNROLL; ++o) {
        v2f a[UNROLL], b[UNROLL];
#pragma unroll
        for (int u = 0; u < UNROLL; ++u) {
            // A: non-temporal b64 (X is streamed once; keep L2 for Weff)
            a[u] = __builtin_nontemporal_load((const v2f*)(ap + u * 4));
            // B: one b64 per step from the pair-interleaved Weff (L2-hot)
            b[u] = *(const v2f*)(bp + u * 64);
        }
        // Unguarded speculative prefetch of the next 128B line of this lane's X row.
        __builtin_prefetch(ap + UNROLL * 4, 0, 3);
#pragma unroll
        for (int u = 0; u < UNROLL; ++u)
            acc = __builtin_amdgcn_wmma_f32_16x16x4_f32(
                /*neg_a=*/false, a[u], /*neg_b=*/false, b[u],
                /*c_mod=*/(short)0, acc, /*reuse_a=*/false, /*reuse_b=*/false);
        ap += UNROLL * 4;
        bp += UNROLL * 64;
    }

    // C/D layout: VGPR r -> M = r + 8*half, N = sub. Reduce K-segments via f32 atomics.
    float* st = stage + tile * 256 + sub;
    const int mbase = half * 8;
#pragma unroll
    for (int r = 0; r < 8; ++r)
        atomicAdd(st + (mbase + r) * 16, acc[r]);
}

__global__ void out_k(const float* __restrict__ stage,
                      const float* __restrict__ bagg,
                      float* __restrict__ out) {
    int t = blockIdx.x * 256 + threadIdx.x;
    if (t >= B_ * C_) return;
    int b = t / C_, c = t % C_;
    out[t] = stage[(b >> 4) * 256 + (b & 15) * 16 + c] + bagg[c];
}

extern "C" void kernel_launch(void* const* d_in, const int* in_sizes, int n_in,
                              void* d_out, int out_size, void* d_ws, size_t ws_size,
                              hipStream_t stream) {
    const float* X     = (const float*)d_in[0];  // (B, G)
    const float* W     = (const float*)d_in[1];  // (L, g, C)
    const float* bias  = (const float*)d_in[2];  // (L, C)
    const float* agg_w = (const float*)d_in[3];  // (1, L)
    const float* agg_b = (const float*)d_in[4];  // (1,)
    const int*   gidx  = (const int*)d_in[5];    // (L, g)

    float* ws    = (float*)d_ws;
    float* weff  = ws + OFF_WEFF;
    float* stage = ws + OFF_STAGE;
    float* bagg  = ws + OFF_BAGG;
    float* out   = (float*)d_out;

    (void)in_sizes; (void)n_in; (void)out_size; (void)ws_size;

    const int zgrid = (int)((WS_FLOATS + 255) / 256);
    zero_ws_k<<<zgrid, 256, 0, stream>>>(ws);
    bias_agg_k<<<1, 32, 0, stream>>>(bias, agg_w, agg_b, bagg);
    scatter_k<<<(L_ * g_ + 255) / 256, 256, 0, stream>>>(W, agg_w, gidx, weff);
    gemm_k<<<TILES_M * KSPLIT, 32, 0, stream>>>(X, weff, stage);
    out_k<<<(B_ * C_ + 255) / 256, 256, 0, stream>>>(stage, bagg, out);
}